// TGCN_10230612099279
// MI455X (gfx1250) — compile-verified
//
#include <hip/hip_runtime.h>
#include <hip/hip_bf16.h>

typedef __attribute__((ext_vector_type(2))) float v2f;
typedef __attribute__((ext_vector_type(8))) float v8f;

#define NNODES 100000
#define TSTEPS 8
#define NEDGES 1000000
#define XDIM   128
#define HDIM   64
#define ZDIM   32

// ---------------------------------------------------------------------------
// Generic fp32 WMMA GEMM:  C[M,Nc] = A[M,K] * B + bias
//   TRANSB==0 : B is [K,Nc] row-major        (B[k,c] = Bp[k*Nc + c])
//   TRANSB==1 : B is [Nc,K] row-major, used transposed (B[k,c] = Bp[c*K + k])
// One wave computes one 16x16 tile with V_WMMA_F32_16X16X4_F32.
// A-frag layout (ISA 7.12.2): lane<16 -> M=lane, K={k,k+1}; lane>=16 -> K={k+2,k+3}
// C/D layout: vgpr j -> M = j + 8*(lane>>4), N = lane&15
// ---------------------------------------------------------------------------
template <int TRANSB>
__global__ __launch_bounds__(256) void wmma_gemm_f32(
    const float* __restrict__ A, const float* __restrict__ B,
    const float* __restrict__ bias, float* __restrict__ C,
    int M, int K, int Nc)
{
    const int wavesPerBlock = blockDim.x >> 5;
    const int wave   = blockIdx.x * wavesPerBlock + (threadIdx.x >> 5);
    const int tilesN = Nc >> 4;
    const int trow   = wave / tilesN;
    const int tcol   = wave % tilesN;
    if (trow * 16 >= M) return;                 // wave-uniform exit: EXEC stays full

    const int lane = threadIdx.x & 31;
    const int m    = lane & 15;
    const int half = lane >> 4;
    const int col  = tcol * 16 + m;

    const float* __restrict__ Arow = A + (size_t)(trow * 16 + m) * K;

    v8f acc = {};
    for (int k = 0; k < K; k += 4) {
        const int ka = k + half * 2;
        v2f a, b;
        a.x = Arow[ka];
        a.y = Arow[ka + 1];
        if (TRANSB) {
            b.x = B[(size_t)col * K + ka];
            b.y = B[(size_t)col * K + ka + 1];
        } else {
            b.x = B[(size_t)ka * Nc + col];
            b.y = B[(size_t)(ka + 1) * Nc + col];
        }
        acc = __builtin_amdgcn_wmma_f32_16x16x4_f32(
            false, a, false, b, (short)0, acc, false, false);
    }

    const float bv = bias ? bias[col] : 0.0f;
    #pragma unroll
    for (int j = 0; j < 8; ++j) {
        const int row = trow * 16 + j + half * 8;
        C[(size_t)row * Nc + col] = acc[j] + bv;
    }
}

// ---------------------------------------------------------------------------
// Elementwise helpers
// ---------------------------------------------------------------------------
__global__ void fill_f32(float* __restrict__ p, float v, long long n) {
    long long i = (long long)blockIdx.x * blockDim.x + threadIdx.x;
    if (i < n) p[i] = v;
}

__global__ void deg_accum(const int* __restrict__ dst, const float* __restrict__ ew,
                          float* __restrict__ deg, int E) {
    int e = blockIdx.x * blockDim.x + threadIdx.x;
    if (e < E) atomicAdd(&deg[dst[e]], ew[e]);
}

__global__ void deg_to_dinv(float* __restrict__ deg, int n) {
    int i = blockIdx.x * blockDim.x + threadIdx.x;
    if (i < n) deg[i] = rsqrtf(deg[i]);
}

// One thread per (edge, feature). 64 consecutive threads share one edge, so the
// dinv/ew/src/dst loads are broadcast within the wave and the H reads / atomic
// adds are fully coalesced 64-float rows.
__global__ void edge_scatter(const int* __restrict__ src, const int* __restrict__ dst,
                             const float* __restrict__ ew, const float* __restrict__ dinv,
                             const float* __restrict__ H, float* __restrict__ out, int E)
{
    long long idx = (long long)blockIdx.x * blockDim.x + threadIdx.x;
    int e = (int)(idx >> 6);
    int f = (int)(idx & 63);
    if (e >= E) return;
    int s = src[e], d = dst[e];
    float norm = dinv[s] * ew[e] * dinv[d];
    atomicAdd(&out[(size_t)d * HDIM + f], norm * H[(size_t)s * HDIM + f]);
}

// out = act(agg + dinv^2 * base)    (dinv^2 == 1/deg, self-loop term)
template <int ACT>  // 0 = relu, 1 = tanh
__global__ void combine_act(const float* __restrict__ agg, const float* __restrict__ base,
                            const float* __restrict__ dinv, float* __restrict__ out,
                            long long n)
{
    long long i = (long long)blockIdx.x * blockDim.x + threadIdx.x;
    if (i >= n) return;
    int node = (int)(i >> 6);
    float inv = dinv[node];
    float v = agg[i] + (inv * inv) * base[i];
    out[i] = (ACT == 0) ? fmaxf(v, 0.0f) : tanhf(v);
}

__device__ __forceinline__ float sigmoidf(float x) { return 1.0f / (1.0f + expf(-x)); }

// PyTorch gate order (r, z, n); gi/gh are [N,192] with biases already added.
__global__ void gru_gate(const float* __restrict__ gi, const float* __restrict__ gh,
                         const float* __restrict__ hprev, float* __restrict__ hnew,
                         long long n)
{
    long long i = (long long)blockIdx.x * blockDim.x + threadIdx.x;
    if (i >= n) return;
    int node = (int)(i >> 6);
    int c    = (int)(i & 63);
    size_t b = (size_t)node * (3 * HDIM) + c;
    float r  = sigmoidf(gi[b]       + gh[b]);
    float z  = sigmoidf(gi[b + 64]  + gh[b + 64]);
    float nn = tanhf   (gi[b + 128] + r * gh[b + 128]);
    hnew[i] = (1.0f - z) * nn + z * hprev[i];
}

// ---------------------------------------------------------------------------
// Host orchestration
// ---------------------------------------------------------------------------
static inline int cdiv(long long a, long long b) { return (int)((a + b - 1) / b); }

extern "C" void kernel_launch(void* const* d_in, const int* in_sizes, int n_in,
                              void* d_out, int out_size, void* d_ws, size_t ws_size,
                              hipStream_t stream)
{
    (void)in_sizes; (void)n_in; (void)out_size; (void)ws_size;

    const float* x     = (const float*)d_in[0];   // [N,128]
    const int*   ei    = (const int*)  d_in[1];   // [T,2,E]
    const float* ew    = (const float*)d_in[2];   // [T,E]
    const float* W1    = (const float*)d_in[3];   // [128,64]
    const float* b1    = (const float*)d_in[4];
    const float* W2    = (const float*)d_in[5];   // [64,64]
    const float* b2    = (const float*)d_in[6];
    const float* W_ih  = (const float*)d_in[7];   // [192,64]
    const float* W_hh  = (const float*)d_in[8];   // [192,64]
    const float* b_ih  = (const float*)d_in[9];
    const float* b_hh  = (const float*)d_in[10];
    const float* W_out = (const float*)d_in[11];  // [64,32]
    const float* b_out = (const float*)d_in[12];
    float* out = (float*)d_out;                   // [T*N*32] ++ [N*64]

    const int Nn = NNODES, T = TSTEPS, E = NEDGES;
    const long long NH  = (long long)Nn * HDIM;   // 6.4M
    const long long NG  = (long long)Nn * 3 * HDIM;

    // workspace layout (floats)
    float* ws   = (float*)d_ws;
    float* XW1  = ws;               // N*64   : x@W1+b1 (t-invariant)
    float* dinv = XW1  + NH;        // N      : deg -> rsqrt(deg)
    float* agg  = dinv + Nn;        // N*64
    float* hbuf = agg  + NH;        // N*64   : relu hidden
    float* HW2  = hbuf + NH;        // N*64
    float* zbuf = HW2  + NH;        // N*64   : tanh output z_t
    float* gi   = zbuf + NH;        // N*192
    float* gh   = gi   + NG;        // N*192
    float* hA   = gh   + NG;        // N*64   : GRU h ping
    float* hB   = hA   + NH;        // N*64   : GRU h pong

    const int TPB = 256;
    const int wavesPB = TPB / 32;

    // ---- t-invariant input projection: XW1 = x @ W1 + b1  (K=128, Nc=64)
    {
        int tiles = (Nn / 16) * (HDIM / 16);
        wmma_gemm_f32<0><<<cdiv(tiles, wavesPB), TPB, 0, stream>>>(x, W1, b1, XW1, Nn, XDIM, HDIM);
    }
    // ---- h0 = 0
    fill_f32<<<cdiv(NH, TPB), TPB, 0, stream>>>(hA, 0.0f, NH);

    float* hprev = hA;
    float* hnext = hB;

    for (int t = 0; t < T; ++t) {
        const int*   src_t = ei + (size_t)t * 2 * E;
        const int*   dst_t = src_t + E;
        const float* ew_t  = ew + (size_t)t * E;

        // degrees (self-loop weight 1) -> dinv = rsqrt(deg)
        fill_f32<<<cdiv(Nn, TPB), TPB, 0, stream>>>(dinv, 1.0f, Nn);
        deg_accum<<<cdiv(E, TPB), TPB, 0, stream>>>(dst_t, ew_t, dinv, E);
        deg_to_dinv<<<cdiv(Nn, TPB), TPB, 0, stream>>>(dinv, Nn);

        // GCN layer 1: scatter norm * XW1[src] into agg, combine + relu
        fill_f32<<<cdiv(NH, TPB), TPB, 0, stream>>>(agg, 0.0f, NH);
        edge_scatter<<<cdiv((long long)E * HDIM, TPB), TPB, 0, stream>>>(
            src_t, dst_t, ew_t, dinv, XW1, agg, E);
        combine_act<0><<<cdiv(NH, TPB), TPB, 0, stream>>>(agg, XW1, dinv, hbuf, NH);

        // GCN layer 2: HW2 = h @ W2 + b2, scatter, combine + tanh -> z_t
        {
            int tiles = (Nn / 16) * (HDIM / 16);
            wmma_gemm_f32<0><<<cdiv(tiles, wavesPB), TPB, 0, stream>>>(hbuf, W2, b2, HW2, Nn, HDIM, HDIM);
        }
        fill_f32<<<cdiv(NH, TPB), TPB, 0, stream>>>(agg, 0.0f, NH);
        edge_scatter<<<cdiv((long long)E * HDIM, TPB), TPB, 0, stream>>>(
            src_t, dst_t, ew_t, dinv, HW2, agg, E);
        combine_act<1><<<cdiv(NH, TPB), TPB, 0, stream>>>(agg, HW2, dinv, zbuf, NH);

        // GRU: gi = z_t @ W_ih^T + b_ih ; gh = h @ W_hh^T + b_hh  (K=64, Nc=192)
        {
            int tiles = (Nn / 16) * ((3 * HDIM) / 16);
            wmma_gemm_f32<1><<<cdiv(tiles, wavesPB), TPB, 0, stream>>>(zbuf,  W_ih, b_ih, gi, Nn, HDIM, 3 * HDIM);
            wmma_gemm_f32<1><<<cdiv(tiles, wavesPB), TPB, 0, stream>>>(hprev, W_hh, b_hh, gh, Nn, HDIM, 3 * HDIM);
        }
        gru_gate<<<cdiv(NH, TPB), TPB, 0, stream>>>(gi, gh, hprev, hnext, NH);

        // out[t] = h_new @ W_out + b_out  (K=64, Nc=32) straight into d_out
        {
            int tiles = (Nn / 16) * (ZDIM / 16);
            wmma_gemm_f32<0><<<cdiv(tiles, wavesPB), TPB, 0, stream>>>(
                hnext, W_out, b_out, out + (size_t)t * Nn * ZDIM, Nn, HDIM, ZDIM);
        }

        float* tmp = hprev; hprev = hnext; hnext = tmp;
    }

    // h_final -> tail of d_out
    hipMemcpyAsync(out + (size_t)T * Nn * ZDIM, hprev, (size_t)NH * sizeof(float),
                   hipMemcpyDeviceToDevice, stream);
}